// VectorQuantizer_1357209666240
// MI455X (gfx1250) — compile-verified
//
#include <hip/hip_runtime.h>

typedef float v2f __attribute__((ext_vector_type(2)));
typedef float v8f __attribute__((ext_vector_type(8)));

#define NUM_EMB   1024
#define EMB_DIM   64
#define SPATIAL   16384                 // 16*32*32
#define BATCH     8
#define N_ROWS    (BATCH * SPATIAL)     // 131072 quantization vectors
#define Z_ELEMS   (BATCH * EMB_DIM * SPATIAL)  // 8388608

// ---------------------------------------------------------------------------
// Kernel 0: bias[k] = -0.5 * ||e_k||^2   (drop the per-row ||z||^2 constant);
// also zero the loss accumulator (d_out is poisoned, never re-poisoned).
// ---------------------------------------------------------------------------
__global__ void vq_bias_kernel(const float* __restrict__ emb,
                               float* __restrict__ bias,
                               float* __restrict__ loss) {
    int k = blockIdx.x * blockDim.x + threadIdx.x;
    if (k < NUM_EMB) {
        const float* e = emb + (size_t)k * EMB_DIM;
        float s = 0.f;
#pragma unroll
        for (int c = 0; c < EMB_DIM; ++c) s += e[c] * e[c];
        bias[k] = -0.5f * s;
    }
    if (blockIdx.x == 0 && threadIdx.x == 0) *loss = 0.f;
}

// ---------------------------------------------------------------------------
// Kernel 1: fused GEMM (z_flat @ emb^T via V_WMMA_F32_16X16X4_F32) + argmax.
// One wave handles a 16-row tile against all 1024 codes; scores never touch
// memory (saves ~1 GB of HBM traffic vs. materializing the distance matrix).
// ---------------------------------------------------------------------------
__global__ void __launch_bounds__(256)
vq_argmax_kernel(const float* __restrict__ z,
                 const float* __restrict__ emb,
                 const float* __restrict__ bias,
                 int* __restrict__ idx_out) {
    const int lane = threadIdx.x & 31;
    const int wave = threadIdx.x >> 5;
    const int tile = blockIdx.x * (blockDim.x >> 5) + wave;   // 0..8191
    const int n0   = tile * 16;                                // first row of tile
    const int b    = n0 / SPATIAL;                             // batch (tiles never straddle)
    const int s0   = n0 % SPATIAL;                             // spatial base
    const int mcol = lane & 15;                                // row (for A) / col (for B,C)
    const int khi  = (lane >> 4) << 1;                         // 0 or 2 (K sub-slot)

    // --- Load A tile: rows n0..n0+15, all 64 channels, ISA fp32 A layout. ---
    // z element (b,c,s) lives at b*EMB_DIM*SPATIAL + c*SPATIAL + s (channel-strided).
    const float* zb = z + (size_t)b * (EMB_DIM * SPATIAL) + (size_t)(s0 + mcol);
    v2f a[16];
#pragma unroll
    for (int kk = 0; kk < 16; ++kk) {
        const int c0 = 4 * kk + khi;
        a[kk].x = zb[(size_t)c0 * SPATIAL];
        a[kk].y = zb[(size_t)(c0 + 1) * SPATIAL];
    }

    float best[8];
    int   bcol[8];
#pragma unroll
    for (int i = 0; i < 8; ++i) { best[i] = -3.0e38f; bcol[i] = 0; }

    // --- Sweep the 1024 codes in 16-column chunks. ---
    for (int j = 0; j < NUM_EMB / 16; ++j) {
        const int   col  = j * 16 + mcol;
        const float bval = bias[col];                 // seed accumulator with -0.5||e||^2
        v8f c = {bval, bval, bval, bval, bval, bval, bval, bval};

        const float* eb = emb + (size_t)col * EMB_DIM + khi;  // row of codebook, contiguous
#pragma unroll
        for (int kk = 0; kk < 16; ++kk) {
            v2f bv = *(const v2f*)(eb + 4 * kk);      // 8B load: channels 4kk+khi, +1
            c = __builtin_amdgcn_wmma_f32_16x16x4_f32(
                    /*neg_a=*/false, a[kk], /*neg_b=*/false, bv,
                    /*c_mod=*/(short)0, c, /*reuse_a=*/false, /*reuse_b=*/false);
        }
        // Running per-lane argmax; strict '>' keeps earliest (lowest) column.
#pragma unroll
        for (int i = 0; i < 8; ++i) {
            if (c[i] > best[i]) { best[i] = c[i]; bcol[i] = col; }
        }
    }

    // --- Reduce across the 16 lanes of each half-wave (cols of the tile). ---
    // C layout: VGPR i holds row i (lanes 0-15) and row i+8 (lanes 16-31).
#pragma unroll
    for (int off = 8; off >= 1; off >>= 1) {
#pragma unroll
        for (int i = 0; i < 8; ++i) {
            float ov = __shfl_xor(best[i], off, 32);  // masks <16: stays in half-wave
            int   oc = __shfl_xor(bcol[i], off, 32);
            if (ov > best[i] || (ov == best[i] && oc < bcol[i])) {
                best[i] = ov; bcol[i] = oc;
            }
        }
    }
    if (lane == 0) {
#pragma unroll
        for (int i = 0; i < 8; ++i) idx_out[n0 + i] = bcol[i];
    } else if (lane == 16) {
#pragma unroll
        for (int i = 0; i < 8; ++i) idx_out[n0 + 8 + i] = bcol[i];
    }
}

// ---------------------------------------------------------------------------
// Kernel 2: gather z_q = emb[idx] (coalesced over spatial dim) + fused
// vq_loss = 1.25 * mean((z_q - z)^2) via block reduction + fp32 HW atomic.
// ---------------------------------------------------------------------------
__global__ void __launch_bounds__(256)
vq_gather_loss_kernel(const float* __restrict__ z,
                      const float* __restrict__ emb,
                      const int* __restrict__ idx,
                      float* __restrict__ zq_out,
                      float* __restrict__ loss) {
    const int i = blockIdx.x * 256 + threadIdx.x;      // grid sized exactly
    const int s = i & (SPATIAL - 1);
    const int c = (i >> 14) & (EMB_DIM - 1);
    const int b = i >> 20;                              // 2^20 = EMB_DIM*SPATIAL
    const int n = b * SPATIAL + s;

    const int   code = idx[n];
    const float e    = emb[(size_t)code * EMB_DIM + c];
    const float zv   = z[i];
    zq_out[i] = e;                                      // z_q_st == z_q numerically
    float d    = zv - e;
    float part = d * d;

#pragma unroll
    for (int off = 16; off >= 1; off >>= 1) part += __shfl_xor(part, off, 32);

    __shared__ float red[8];
    if ((threadIdx.x & 31) == 0) red[threadIdx.x >> 5] = part;
    __syncthreads();
    if (threadIdx.x == 0) {
        float bs = 0.f;
#pragma unroll
        for (int w = 0; w < 8; ++w) bs += red[w];
        // vq_loss = codebook + 0.25*commitment = 1.25 * mean
        unsafeAtomicAdd(loss, bs * (1.25f / (float)Z_ELEMS));
    }
}

// ---------------------------------------------------------------------------
extern "C" void kernel_launch(void* const* d_in, const int* in_sizes, int n_in,
                              void* d_out, int out_size, void* d_ws, size_t ws_size,
                              hipStream_t stream) {
    const float* z   = (const float*)d_in[0];   // (8, 64, 16, 32, 32) fp32
    const float* emb = (const float*)d_in[1];   // (1024, 64) fp32

    float* out  = (float*)d_out;
    float* zq   = out;                          // 8388608 floats
    float* loss = out + Z_ELEMS;                // 1 float
    int*   idx  = (int*)(out + Z_ELEMS + 1);    // 131072 int32

    float* bias = (float*)d_ws;                 // 1024 floats scratch

    vq_bias_kernel<<<4, 256, 0, stream>>>(emb, bias, loss);
    vq_argmax_kernel<<<(N_ROWS / 16) / 8, 256, 0, stream>>>(z, emb, bias, idx);
    vq_gather_loss_kernel<<<Z_ELEMS / 256, 256, 0, stream>>>(z, emb, idx, zq, loss);
}